// Combine_Conv2d_34565896799013
// MI455X (gfx1250) — compile-verified
//
#include <hip/hip_runtime.h>

// ---------------------------------------------------------------------------
// Distance-conv + Gaussian RBF for MI455X (gfx1250), f32 WMMA path.
//   d2 = sum(p^2) - 2 p.w + sum(w^2)   (cross term = GEMM via V_WMMA_F32_16X16X4_F32)
//   out = exp(-(sqrt(d2_g)+sqrt(d2_r))^2 / (2 std^2))
// ---------------------------------------------------------------------------

typedef __attribute__((ext_vector_type(2))) float v2f;
typedef __attribute__((ext_vector_type(8))) float v8f;

#define BATCH   16
#define NFILT   256
#define KSZ     5
#define KK      25          // 5x5
#define KSTEPS  7           // K padded 25 -> 28, 7 steps of 4
#define HIN     112
#define WIN     112
#define HW_IN   (HIN * WIN)
#define HOUT    108
#define WOUT    108
#define HW_OUT  (HOUT * WOUT)   // 11664 (divisible by 16)

// --- pre-kernel: ||w||^2 per filter, both convs, into workspace ------------
__global__ void w2_precompute(const float* __restrict__ wg,
                              const float* __restrict__ wr,
                              float* __restrict__ w2ws) {
    int t = blockIdx.x * blockDim.x + threadIdx.x;   // 0..511
    if (t >= 2 * NFILT) return;
    const float* w = (t < NFILT) ? wg : wr;
    int n = t & (NFILT - 1);
    float s = 0.0f;
#pragma unroll
    for (int k = 0; k < KK; ++k) {
        float v = w[n * KK + k];
        s = fmaf(v, v, s);
    }
    w2ws[t] = s;
}

// --- main kernel: one wave32 per 16-pixel M-tile ---------------------------
__global__ __launch_bounds__(256) void dist_rbf_wmma(
    const float* __restrict__ xg, const float* __restrict__ xr,
    const float* __restrict__ wg, const float* __restrict__ wr,
    const float* __restrict__ stdp, const float* __restrict__ w2ws,
    float* __restrict__ out)
{
    const int lane = threadIdx.x & 31;
    const int m    = lane & 15;      // A-row / B-col / C-col index
    const int hi   = lane >> 4;      // lane half (K offset +2, C rows +8)
    const int tile = blockIdx.x * (blockDim.x >> 5) + (threadIdx.x >> 5);

    // pixel handled by this lane (A-matrix row m)
    const int p  = tile * 16 + m;
    const int b  = p / HW_OUT;               // image (same for whole wave)
    const int hw = p - b * HW_OUT;
    const int h  = hw / WOUT;
    const int w  = hw - h * WOUT;
    const float* xgb = xg + b * HW_IN + h * WIN + w;
    const float* xrb = xr + b * HW_IN + h * WIN + w;

    // ---- A fragments for all 7 K-steps, both convs (reused over 16 N-tiles)
    v2f ag[KSTEPS], ar[KSTEPS];
#pragma unroll
    for (int s = 0; s < KSTEPS; ++s) {
#pragma unroll
        for (int j = 0; j < 2; ++j) {
            int k = 4 * s + j + 2 * hi;      // ISA 32-bit A layout
            float vg = 0.0f, vr = 0.0f;
            if (k < KK) {
                int kh = k / KSZ, kw = k - kh * KSZ;
                vg = xgb[kh * WIN + kw];
                vr = xrb[kh * WIN + kw];
            }
            ag[s][j] = vg;
            ar[s][j] = vr;
        }
    }

    // ---- per-pixel sum of squares over the 5x5 window (lane m -> pixel m)
    float x2g = 0.0f, x2r = 0.0f;
#pragma unroll
    for (int k = 0; k < KK; ++k) {
        int kh = k / KSZ, kw = k - kh * KSZ;
        float vg = xgb[kh * WIN + kw];
        float vr = xrb[kh * WIN + kw];
        x2g = fmaf(vg, vg, x2g);
        x2r = fmaf(vr, vr, x2r);
    }
    // redistribute to C-layout rows: register r holds M = r + 8*hi
    float x2gm[8], x2rm[8];
#pragma unroll
    for (int r = 0; r < 8; ++r) {
        int src = r + 8 * hi;
        x2gm[r] = __shfl(x2g, src, 32);
        x2rm[r] = __shfl(x2r, src, 32);
    }

    const float stdv  = stdp[0];
    const float scale = -1.0f / (2.0f * stdv * stdv);

    // store geometry: registers r=0..7 are pixels (tile*16 + 8*hi) + r
    const int hwbase = hw - m + 8 * hi;      // multiple of 8 -> 32B aligned
    float* outb = out + (size_t)b * NFILT * HW_OUT;

    for (int nt = 0; nt < NFILT / 16; ++nt) {
        const int n = nt * 16 + m;           // B-matrix column for this lane
        v8f accg = {};
        v8f accr = {};
#pragma unroll
        for (int s = 0; s < KSTEPS; ++s) {
            v2f bg, br;
#pragma unroll
            for (int j = 0; j < 2; ++j) {
                int k = 4 * s + j + 2 * hi;  // ISA 32-bit B layout (mirrors A)
                float vg = 0.0f, vr = 0.0f;
                if (k < KK) {
                    vg = wg[n * KK + k];
                    vr = wr[n * KK + k];
                }
                bg[j] = vg;
                br[j] = vr;
            }
            accg = __builtin_amdgcn_wmma_f32_16x16x4_f32(
                false, ag[s], false, bg, (short)0, accg, false, false);
            accr = __builtin_amdgcn_wmma_f32_16x16x4_f32(
                false, ar[s], false, br, (short)0, accr, false, false);
        }

        const float w2g = w2ws[n];
        const float w2r = w2ws[NFILT + n];

        float res[8];
#pragma unroll
        for (int r = 0; r < 8; ++r) {
            float d2g = x2gm[r] - 2.0f * accg[r] + w2g;
            float d2r = x2rm[r] - 2.0f * accr[r] + w2r;
            // raw v_sqrt_f32: inputs clamped >= 1e-12, ~1 ulp is plenty here
            float d = __builtin_amdgcn_sqrtf(fmaxf(d2g, 1e-12f)) +
                      __builtin_amdgcn_sqrtf(fmaxf(d2r, 1e-12f));
            res[r] = __expf(d * d * scale);
        }

        float4* dst = (float4*)(outb + (size_t)n * HW_OUT + hwbase);
        dst[0] = make_float4(res[0], res[1], res[2], res[3]);
        dst[1] = make_float4(res[4], res[5], res[6], res[7]);
    }
}

extern "C" void kernel_launch(void* const* d_in, const int* in_sizes, int n_in,
                              void* d_out, int out_size, void* d_ws, size_t ws_size,
                              hipStream_t stream) {
    const float* gray_in  = (const float*)d_in[0];
    const float* rgb_in   = (const float*)d_in[1];
    const float* gray_w   = (const float*)d_in[2];
    const float* rgb_w    = (const float*)d_in[3];
    const float* std_p    = (const float*)d_in[4];
    float*       out      = (float*)d_out;
    float*       w2ws     = (float*)d_ws;    // 512 floats

    // ||w||^2 per (filter, conv)
    w2_precompute<<<2, 256, 0, stream>>>(gray_w, rgb_w, w2ws);

    // one wave32 per 16-pixel tile: BATCH*HW_OUT/16 = 11664 waves, 8 per block
    const int total_waves = BATCH * HW_OUT / 16;     // 11664
    const int waves_per_block = 8;                   // 256 threads
    dim3 grid(total_waves / waves_per_block);        // 1458
    dim3 block(256);
    dist_rbf_wmma<<<grid, block, 0, stream>>>(gray_in, rgb_in, gray_w, rgb_w,
                                              std_p, w2ws, out);
}